// FernSparseTable_37512244364034
// MI455X (gfx1250) — compile-verified
//
#include <hip/hip_runtime.h>
#include <stdint.h>

// FernSparseTable on gfx1250:
//  - per-fern 256KB vote table staged into LDS via 8 parallel TDM descriptors
//    (one per wave, 32KB slice each), overlapped with the per-pixel ALU phase
//  - conflict-free d-major LDS gathers (ds_load_b128) + f32 FMA accumulation
//  - non-temporal policy for the single-use B / out streams (keeps tables in L2)

#define FERNS      16
#define KB         10
#define PPB        256                 // pixels per block
#define TAB_FLOATS (1024 * 64)         // one fern table: 256 KB
#define SEG_FLOATS (TAB_FLOATS / 8)    // per-wave DMA slice: 8192 floats = 32 KB
#define SMEM_BYTES (TAB_FLOATS * 4 + PPB * 8 * 4 + PPB * 8 * 4)   // 272 KB

typedef unsigned int u32x4 __attribute__((ext_vector_type(4)));
typedef int          i32x8 __attribute__((ext_vector_type(8)));
typedef int          i32x4 __attribute__((ext_vector_type(4)));
typedef float        f32x4 __attribute__((ext_vector_type(4)));

__global__ __launch_bounds__(256)
void fern_sparse_table_kernel(const float* __restrict__ B,
                              const float* __restrict__ tables,
                              float* __restrict__ out)
{
    extern __shared__ char smem[];
    float*    tab  = (float*)smem;                                   // 256 KB
    uint32_t* sIdx = (uint32_t*)(smem + TAB_FLOATS * 4);             // 8 KB
    float*    sW   = (float*)(smem + TAB_FLOATS * 4 + PPB * 8 * 4);  // 8 KB

    const int t  = threadIdx.x;
    const int p  = blockIdx.x * PPB + t;          // global pixel id
    const int n  = p >> 10;                       // batch index (H*W = 1024)
    const int hw = p & 1023;
    const float* bpix = B + (size_t)n * (FERNS * KB * 1024) + hw;

    const int dq   = t & 15;                      // d-quad: d = 4*dq .. 4*dq+3
    const int prow = t >> 4;                      // pixel slot 0..15

    // per-wave DMA slice (wave-uniform)
    const int      seg      = t >> 5;             // 0..7
    const uint32_t lds_base = (uint32_t)(uintptr_t)tab + (uint32_t)seg * (SEG_FLOATS * 4);
    const uint64_t ga_base  = (uint64_t)(uintptr_t)(tables + (size_t)seg * SEG_FLOATS);

    f32x4 acc[PPB / 16];
#pragma unroll
    for (int i = 0; i < PPB / 16; ++i) acc[i] = (f32x4){0.f, 0.f, 0.f, 0.f};

    for (int f = 0; f < FERNS; ++f) {
        __syncthreads();   // previous iteration's readers done before LDS reuse

        // ---- stage this fern's table into LDS: 8 TDM descriptors, 1/wave ----
        {
            uint64_t ga = ga_base + (uint64_t)f * (TAB_FLOATS * 4);
            u32x4 g0;
            g0[0] = 1u;                               // count=1 (valid descriptor)
            g0[1] = lds_base;                         // lds_addr (byte offset)
            g0[2] = (uint32_t)ga;                     // global_addr[31:0]
            g0[3] = (uint32_t)((ga >> 32) & 0x01FFFFFFu) | 0x80000000u; // addr[56:32] | type=2
            i32x8 g1;
            g1[0] = (int)(2u << 16);                  // data_size = 4 bytes
            g1[1] = (int)((uint32_t)SEG_FLOATS << 16);// tensor_dim0 = 8192 (lo16)
            g1[2] = (int)(1u << 16);                  // tensor_dim0 hi=0 | tensor_dim1 = 1
            g1[3] = (int)((uint32_t)SEG_FLOATS << 16);// tensor_dim1 hi=0 | tile_dim0 = 8192
            g1[4] = 1;                                // tile_dim1 = 1, tile_dim2 = 0
            g1[5] = SEG_FLOATS;                       // tensor_dim0_stride (elements)
            g1[6] = 0;
            g1[7] = 0;
            i32x4 gz4 = {0, 0, 0, 0};
            i32x8 gz8 = {0, 0, 0, 0, 0, 0, 0, 0};
            __builtin_amdgcn_tensor_load_to_lds(g0, g1, gz4, gz4, gz8, 0);
        }

        // ---- per-pixel fern codes: base word + 3 most-ambiguous bits ----
        {
            float T[KB], ba[KB];
            const float* c = bpix + f * KB * 1024;
#pragma unroll
            for (int k = 0; k < KB; ++k) T[k] = __builtin_nontemporal_load(c + k * 1024);
            int wb = 0;
#pragma unroll
            for (int k = 0; k < KB; ++k) {
                wb |= (T[k] > 0.5f ? 1 : 0) << (KB - 1 - k);  // round-half-even on [0,1)
                ba[k] = fabsf(T[k] - 0.5f);
            }
            int masks[3]; float p1[3];
#pragma unroll
            for (int j = 0; j < 3; ++j) {
                int best = 0; float bv = ba[0];
#pragma unroll
                for (int k = 1; k < KB; ++k)
                    if (ba[k] < bv) { bv = ba[k]; best = k; }  // first-min == jnp.argmin
                p1[j] = T[best];
                ba[best] += 1.0f;                              // exclude from later argmins
                masks[j] = 1 << (KB - 1 - best);
            }
            const int base_it = wb & ~(masks[0] | masks[1] | masks[2]);
#pragma unroll
            for (int a = 0; a < 8; ++a) {
                float w =  ( a       & 1) ? p1[0] : 1.0f - p1[0];
                w      *= (((a >> 1) & 1) ? p1[1] : 1.0f - p1[1]);
                w      *= (((a >> 2) & 1) ? p1[2] : 1.0f - p1[2]);
                int it = base_it;
                if ( a       & 1) it |= masks[0];
                if ((a >> 1) & 1) it |= masks[1];
                if ((a >> 2) & 1) it |= masks[2];
                sIdx[t * 8 + a] = (uint32_t)it;
                sW  [t * 8 + a] = w;
            }
        }

        __builtin_amdgcn_s_wait_tensorcnt(0);   // each wave waits on its own DMA
        __syncthreads();

        // ---- gather-accumulate from LDS (d-major lanes: conflict-free b128) ----
#pragma unroll
        for (int ppi = 0; ppi < PPB / 16; ++ppi) {
            const int pp = prow + ppi * 16;
            f32x4 a4 = acc[ppi];
#pragma unroll
            for (int a = 0; a < 8; ++a) {
                const uint32_t row = sIdx[pp * 8 + a];
                const float    w   = sW  [pp * 8 + a];
                const f32x4    v   = *(const f32x4*)(tab + row * 64 + dq * 4);
                a4.x = fmaf(w, v.x, a4.x);
                a4.y = fmaf(w, v.y, a4.y);
                a4.z = fmaf(w, v.z, a4.z);
                a4.w = fmaf(w, v.w, a4.w);
            }
            acc[ppi] = a4;
        }
    }

    // ---- coalesced non-temporal float4 output stores: out[p*64 + d] ----
    float* o = out + (size_t)blockIdx.x * PPB * 64;
#pragma unroll
    for (int ppi = 0; ppi < PPB / 16; ++ppi) {
        const int pp = prow + ppi * 16;
        __builtin_nontemporal_store(acc[ppi], (f32x4*)(o + pp * 64 + dq * 4));
    }
}

extern "C" void kernel_launch(void* const* d_in, const int* in_sizes, int n_in,
                              void* d_out, int out_size, void* d_ws, size_t ws_size,
                              hipStream_t stream) {
    const float* B      = (const float*)d_in[0];   // [64,160,32,32] f32
    const float* tables = (const float*)d_in[1];   // [16,1024,64] f32
    float*       out    = (float*)d_out;           // [64,32,32,64] f32

    const int npix = 64 * 32 * 32;
    dim3 grid(npix / PPB), block(256);
    fern_sparse_table_kernel<<<grid, block, SMEM_BYTES, stream>>>(B, tables, out);
}